// HyperGCNBlock_51196010168978
// MI455X (gfx1250) — compile-verified
//
#include <hip/hip_runtime.h>
#include <math.h>

typedef float v2f __attribute__((ext_vector_type(2)));
typedef float v8f __attribute__((ext_vector_type(8)));

#if defined(__AMDGCN__) && __has_builtin(__builtin_amdgcn_wmma_f32_16x16x4_f32)
#define USE_WMMA_F32 1
#else
#define USE_WMMA_F32 0
#endif

#define DIMF 128
#define NGRAPH 32

static inline int cdiv_i(long long a, long long b) { return (int)((a + b - 1) / b); }

// ---------------------------------------------------------------- fill
__global__ void fill_f32(float* __restrict__ p, float v, long long n) {
    long long i = (long long)blockIdx.x * blockDim.x + threadIdx.x;
    if (i < n) p[i] = v;
}

// ---------------------------------------------------------------- GEMM: out[R,128] = A[R,K] @ W[K,128] (+ bias)
// 256 threads = 8 waves; wave w -> column tile w*16; block -> 16 rows.
__global__ __launch_bounds__(256) void gemm128_wmma(
    const float* __restrict__ A, const float* __restrict__ W,
    const float* __restrict__ bias, float* __restrict__ out, int R, int K)
{
    const int wave = threadIdx.x >> 5;
    const int lane = threadIdx.x & 31;
    const int half = lane >> 4;       // 0 or 1
    const int l16  = lane & 15;
    const int row0 = blockIdx.x * 16;
    const int col0 = wave * 16;
    const int n    = col0 + l16;      // output column for this lane
    v8f c = {0.f,0.f,0.f,0.f,0.f,0.f,0.f,0.f};
#if USE_WMMA_F32
    int m = row0 + l16; if (m > R - 1) m = R - 1;   // A row (sizes are multiples of 16)
    for (int k0 = 0; k0 < K; k0 += 4) {
        const int ka = k0 + half * 2;
        v2f a, b;
        a.x = A[(long long)m * K + ka];
        a.y = A[(long long)m * K + ka + 1];
        b.x = W[(long long)ka * DIMF + n];
        b.y = W[(long long)(ka + 1) * DIMF + n];
        c = __builtin_amdgcn_wmma_f32_16x16x4_f32(false, a, false, b, (short)0, c, false, false);
    }
#else
    for (int v = 0; v < 8; ++v) {
        int row = row0 + v + half * 8; if (row > R - 1) row = R - 1;
        float acc = 0.f;
        for (int k = 0; k < K; ++k) acc += A[(long long)row * K + k] * W[(long long)k * DIMF + n];
        c[v] = acc;
    }
#endif
    const float bval = bias ? bias[n] : 0.f;
#pragma unroll
    for (int v = 0; v < 8; ++v) {
        const int row = row0 + v + half * 8;
        if (row < R) out[(long long)row * DIMF + n] = c[v] + bval;
    }
}

// ---------------------------------------------------------------- Gram: G[128,128] += Z1^T @ Z2 (split-K, atomics)
__global__ __launch_bounds__(256) void corr_gram_wmma(
    const float* __restrict__ Z1, const float* __restrict__ Z2,
    float* __restrict__ G, int R, int rowsPerBlock)
{
    const int wave = threadIdx.x >> 5;    // M tile (row of G)
    const int lane = threadIdx.x & 31;
    const int half = lane >> 4;
    const int l16  = lane & 15;
    const int nT   = blockIdx.y;          // N tile (col of G)
    const int j    = nT * 16 + l16;       // Z2 column
    const int rbeg = blockIdx.x * rowsPerBlock;
    int rend = rbeg + rowsPerBlock; if (rend > R) rend = R;
    v8f c = {0.f,0.f,0.f,0.f,0.f,0.f,0.f,0.f};
#if USE_WMMA_F32
    const int i = wave * 16 + l16;        // Z1 column (A's M index)
    for (int r = rbeg; r < rend; r += 4) {
        const int ra = r + half * 2;
        v2f a, b;
        a.x = (ra     < R) ? Z1[(long long)ra * DIMF + i]       : 0.f;
        a.y = (ra + 1 < R) ? Z1[(long long)(ra + 1) * DIMF + i] : 0.f;
        b.x = (ra     < R) ? Z2[(long long)ra * DIMF + j]       : 0.f;
        b.y = (ra + 1 < R) ? Z2[(long long)(ra + 1) * DIMF + j] : 0.f;
        c = __builtin_amdgcn_wmma_f32_16x16x4_f32(false, a, false, b, (short)0, c, false, false);
    }
#else
    for (int v = 0; v < 8; ++v) {
        const int gi = wave * 16 + v + half * 8;
        float acc = 0.f;
        for (int r = rbeg; r < rend; ++r)
            acc += Z1[(long long)r * DIMF + gi] * Z2[(long long)r * DIMF + j];
        c[v] = acc;
    }
#endif
#pragma unroll
    for (int v = 0; v < 8; ++v) {
        const int gi = wave * 16 + v + half * 8;
        atomicAdd(&G[gi * DIMF + nT * 16 + l16], c[v]);
    }
}

// ---------------------------------------------------------------- incidence pass A: counts + edge-coord accumulation
__global__ void incidence_count(const int* __restrict__ ni, const int* __restrict__ ei,
                                const float* __restrict__ coord,
                                float* __restrict__ cnt_ni, float* __restrict__ cnt_ei,
                                float* __restrict__ ec_acc, int NNZ)
{
    int t = blockIdx.x * blockDim.x + threadIdx.x;
    if (t >= NNZ) return;
    const int n = ni[t], m = ei[t];
    atomicAdd(&cnt_ni[n], 1.f);
    atomicAdd(&cnt_ei[m], 1.f);
#pragma unroll
    for (int c = 0; c < 3; ++c) atomicAdd(&ec_acc[m * 3 + c], coord[(long long)n * 3 + c]);
}

__global__ void div3_finalize(float* __restrict__ acc, const float* __restrict__ cnt, int n3) {
    int i = blockIdx.x * blockDim.x + threadIdx.x;
    if (i < n3) acc[i] /= fmaxf(cnt[i / 3], 1.f);
}

// ---------------------------------------------------------------- incidence pass B: bipartite scatter + coord message
__global__ __launch_bounds__(128) void incidence_scatter(
    const float* __restrict__ xw, const float* __restrict__ ew,
    const float* __restrict__ coord, const float* __restrict__ edge_coord,
    const int* __restrict__ ni, const int* __restrict__ ei,
    float* __restrict__ e_conv, float* __restrict__ z_conv,
    float* __restrict__ coord_acc, int NNZ)
{
    __shared__ float red[DIMF];
    const int t = blockIdx.x;
    if (t >= NNZ) return;
    const int n = ni[t], m = ei[t];
    const int f = threadIdx.x;
    const float xv = xw[(long long)n * DIMF + f];
    const float ev = ew[(long long)m * DIMF + f];
    atomicAdd(&e_conv[(long long)m * DIMF + f], xv);
    atomicAdd(&z_conv[(long long)n * DIMF + f], ev);
    red[f] = xv * ev;
    __syncthreads();
    for (int s = 64; s > 0; s >>= 1) { if (f < s) red[f] += red[f + s]; __syncthreads(); }
    if (f < 3) {
        const float phi = tanhf(red[0] * 0.08838834764831845f);  // rsqrt(128)
        const float rel = coord[(long long)n * 3 + f] - edge_coord[(long long)m * 3 + f];
        atomicAdd(&coord_acc[(long long)n * 3 + f], rel * phi);
    }
}

__global__ void rowscale(float* __restrict__ Z, const float* __restrict__ cnt, long long total) {
    long long i = (long long)blockIdx.x * blockDim.x + threadIdx.x;
    if (i < total) Z[i] /= fmaxf(cnt[i >> 7], 1.f);
}

__global__ void coord_finalize(const float* __restrict__ coord, const float* __restrict__ cacc,
                               const float* __restrict__ cnt, float* __restrict__ out, int n3) {
    int i = blockIdx.x * blockDim.x + threadIdx.x;
    if (i < n3) out[i] = coord[i] + cacc[i] / fmaxf(cnt[i / 3], 1.f);
}

// ---------------------------------------------------------------- GCN
__global__ void degree_accum(const int* __restrict__ dst, float* __restrict__ deg, int E) {
    int e = blockIdx.x * blockDim.x + threadIdx.x;
    if (e < E) atomicAdd(&deg[dst[e]], 1.f);
}

__global__ __launch_bounds__(128) void gcn_scatter(
    const float* __restrict__ xw, const int* __restrict__ src, const int* __restrict__ dst,
    const float* __restrict__ deg, float* __restrict__ out, int E)
{
    const int e = blockIdx.x;
    if (e >= E) return;
    const int s = src[e], d = dst[e];
    const float norm = rsqrtf(deg[s] * deg[d]);
    atomicAdd(&out[(long long)d * DIMF + threadIdx.x], norm * xw[(long long)s * DIMF + threadIdx.x]);
}

__global__ void gcn_finalize(float* __restrict__ out, const float* __restrict__ xw,
                             const float* __restrict__ deg, const float* __restrict__ bias,
                             long long total) {
    long long i = (long long)blockIdx.x * blockDim.x + threadIdx.x;
    if (i < total) out[i] += xw[i] / deg[i >> 7] + bias[i & (DIMF - 1)];
}

// ---------------------------------------------------------------- column stats (sum / sumsq)
__global__ __launch_bounds__(128) void colstats(const float* __restrict__ Z, int R,
                                                float* __restrict__ sum, float* __restrict__ sq) {
    const int f = threadIdx.x;
    const int r0 = blockIdx.x * 256;
    int r1 = r0 + 256; if (r1 > R) r1 = R;
    float s = 0.f, q = 0.f;
    for (int r = r0; r < r1; ++r) { const float v = Z[(long long)r * DIMF + f]; s += v; q += v * v; }
    atomicAdd(&sum[f], s); atomicAdd(&sq[f], q);
}

// ---------------------------------------------------------------- Barlow loss from Gram + stats
__global__ __launch_bounds__(256) void barlow_loss(
    const float* __restrict__ G, const float* __restrict__ s1, const float* __restrict__ q1,
    const float* __restrict__ s2, const float* __restrict__ q2, int R, float* __restrict__ loss)
{
    __shared__ float red[256];
    const float Rf = (float)R;
    const float ratio = (Rf - 1.f) / Rf;
    float local = 0.f;
    for (int idx = threadIdx.x; idx < DIMF * DIMF; idx += 256) {
        const int i = idx >> 7, j = idx & (DIMF - 1);
        const float m1 = s1[i] / Rf, m2 = s2[j] / Rf;
        const float var1 = (q1[i] - Rf * m1 * m1) / (Rf - 1.f);
        const float var2 = (q2[j] - Rf * m2 * m2) / (Rf - 1.f);
        const float a1 = rsqrtf(var1) * rsqrtf(ratio + 1e-5f);
        const float a2 = rsqrtf(var2) * rsqrtf(ratio + 1e-5f);
        const float Cij = (G[idx] - Rf * m1 * m2) * a1 * a2 * (1.f / 32.f);
        if (i == j) { const float dd = Cij - 1.f; local += dd * dd; }
        else          local += 0.005f * Cij * Cij;
    }
    red[threadIdx.x] = local; __syncthreads();
    for (int s = 128; s > 0; s >>= 1) { if (threadIdx.x < s) red[threadIdx.x] += red[threadIdx.x + s]; __syncthreads(); }
    if (threadIdx.x == 0) atomicAdd(loss, red[0]);
}

// ---------------------------------------------------------------- LayerNorm -> exact GELU -> residual, + graph pooling
__global__ __launch_bounds__(128) void ln_gelu_res_pool(
    const float* __restrict__ conv, const float* __restrict__ resid,
    const float* __restrict__ g, const float* __restrict__ b,
    const int* __restrict__ batch_idx, float* __restrict__ out,
    float* __restrict__ poolsum, float* __restrict__ poolcnt, int R)
{
    __shared__ float red[DIMF];
    const int r = blockIdx.x;
    if (r >= R) return;
    const int f = threadIdx.x;
    const float v = conv[(long long)r * DIMF + f];
    red[f] = v; __syncthreads();
    for (int s = 64; s > 0; s >>= 1) { if (f < s) red[f] += red[f + s]; __syncthreads(); }
    const float mean = red[0] * (1.f / DIMF);
    __syncthreads();
    const float dv = v - mean;
    red[f] = dv * dv; __syncthreads();
    for (int s = 64; s > 0; s >>= 1) { if (f < s) red[f] += red[f + s]; __syncthreads(); }
    const float var = red[0] * (1.f / DIMF);
    const float t = g[f] * dv * rsqrtf(var + 1e-5f) + b[f];
    const float ge = 0.5f * t * (1.f + erff(t * 0.7071067811865476f));
    const float z = ge + resid[(long long)r * DIMF + f];
    out[(long long)r * DIMF + f] = z;
    const int bi = batch_idx[r];
    atomicAdd(&poolsum[bi * DIMF + f], z);
    if (f == 0) atomicAdd(&poolcnt[bi], 1.f);
}

__global__ void pool_finalize(const float* __restrict__ zsum, const float* __restrict__ esum,
                              const float* __restrict__ zc, const float* __restrict__ ec,
                              float* __restrict__ pooled) {
    int idx = blockIdx.x * blockDim.x + threadIdx.x;
    if (idx >= NGRAPH * DIMF) return;
    const int b = idx >> 7, f = idx & (DIMF - 1);
    pooled[b * 2 * DIMF + f]        = zsum[idx] / fmaxf(zc[b], 1.f);
    pooled[b * 2 * DIMF + DIMF + f] = esum[idx] / fmaxf(ec[b], 1.f);
}

// ================================================================ host
extern "C" void kernel_launch(void* const* d_in, const int* in_sizes, int n_in,
                              void* d_out, int out_size, void* d_ws, size_t ws_size,
                              hipStream_t stream) {
    const float* x     = (const float*)d_in[0];
    const float* he    = (const float*)d_in[1];
    const float* coord = (const float*)d_in[2];
    const int*   ni    = (const int*)d_in[3];
    const int*   ei    = (const int*)d_in[4];
    const int*   nb    = (const int*)d_in[5];
    const int*   eb    = (const int*)d_in[6];
    const int*   XX    = (const int*)d_in[7];
    const int*   EEp   = (const int*)d_in[8];
    const float* Wv  = (const float*)d_in[10]; const float* bv  = (const float*)d_in[11];
    const float* We  = (const float*)d_in[12]; const float* be  = (const float*)d_in[13];
    const float* Wx  = (const float*)d_in[14]; const float* bx  = (const float*)d_in[15];
    const float* Weg = (const float*)d_in[16]; const float* beg = (const float*)d_in[17];
    const float* lng = (const float*)d_in[18]; const float* lnb = (const float*)d_in[19];
    const float* Wf  = (const float*)d_in[20]; const float* bf  = (const float*)d_in[21];

    const int N   = in_sizes[0] / DIMF;
    const int M   = in_sizes[1] / DIMF;
    const int NNZ = in_sizes[3];
    const int EG  = in_sizes[7] / 2;
    const int EE  = in_sizes[8] / 2;

    // output layout
    float* out = (float*)d_out;
    float* out_z = out;
    float* out_e = out + (long long)N * DIMF;
    float* out_c = out_e + (long long)M * DIMF;
    float* out_g = out_c + (long long)N * 3;
    float* out_l = out_g + NGRAPH * DIMF;

    // workspace layout (floats)
    float* w = (float*)d_ws;
    size_t o = 0;
    float* buf_xw   = w + o; o += (size_t)N * DIMF;   // xw, later reused for x@Wx
    float* buf_ew   = w + o; o += (size_t)M * DIMF;   // ew, later reused for he@Weg
    float* zconv    = w + o; o += (size_t)N * DIMF;
    float* econv    = w + o; o += (size_t)M * DIMF;
    float* zimp     = w + o; o += (size_t)N * DIMF;
    float* eimp     = w + o; o += (size_t)M * DIMF;
    float* cnt_ni   = w + o; o += (size_t)N;
    float* cnt_ei   = w + o; o += (size_t)M;
    float* ecoord   = w + o; o += (size_t)M * 3;
    float* cacc     = w + o; o += (size_t)N * 3;
    float* deg_x    = w + o; o += (size_t)N;
    float* deg_e    = w + o; o += (size_t)M;
    float* st       = w + o; o += 8 * DIMF;           // col sums/sumsq: zimp,zconv,eimp,econv
    float* G1       = w + o; o += DIMF * DIMF;
    float* G2       = w + o; o += DIMF * DIMF;
    float* poolz    = w + o; o += NGRAPH * DIMF;
    float* poole    = w + o; o += NGRAPH * DIMF;
    float* pcz      = w + o; o += NGRAPH;
    float* pce      = w + o; o += NGRAPH;
    float* pooled   = w + o; o += NGRAPH * 2 * DIMF;
    (void)ws_size; (void)n_in; (void)out_size;

    auto fill = [&](float* p, float v, long long n) {
        fill_f32<<<cdiv_i(n, 256), 256, 0, stream>>>(p, v, n);
    };

    // phase 0: zero/seed accumulators
    fill(zconv, 0.f, (long long)N * DIMF);
    fill(econv, 0.f, (long long)M * DIMF);
    fill(zimp,  0.f, (long long)N * DIMF);
    fill(eimp,  0.f, (long long)M * DIMF);
    fill(cnt_ni, 0.f, N); fill(cnt_ei, 0.f, M);
    fill(ecoord, 0.f, (long long)M * 3); fill(cacc, 0.f, (long long)N * 3);
    fill(deg_x, 1.f, N); fill(deg_e, 1.f, M);
    fill(st, 0.f, 8 * DIMF);
    fill(G1, 0.f, DIMF * DIMF); fill(G2, 0.f, DIMF * DIMF);
    fill(poolz, 0.f, NGRAPH * DIMF); fill(poole, 0.f, NGRAPH * DIMF);
    fill(pcz, 0.f, NGRAPH); fill(pce, 0.f, NGRAPH);
    fill(out_l, 0.f, 1);

    // phase 1: dense projections (WMMA f32 16x16x4)
    gemm128_wmma<<<cdiv_i(N, 16), 256, 0, stream>>>(x,  Wv, bv, buf_xw, N, DIMF);
    gemm128_wmma<<<cdiv_i(M, 16), 256, 0, stream>>>(he, We, be, buf_ew, M, DIMF);

    // phase 2-3: incidence counts, edge coords, bipartite scatter
    incidence_count<<<cdiv_i(NNZ, 256), 256, 0, stream>>>(ni, ei, coord, cnt_ni, cnt_ei, ecoord, NNZ);
    div3_finalize<<<cdiv_i((long long)M * 3, 256), 256, 0, stream>>>(ecoord, cnt_ei, M * 3);
    incidence_scatter<<<NNZ, DIMF, 0, stream>>>(buf_xw, buf_ew, coord, ecoord, ni, ei,
                                                econv, zconv, cacc, NNZ);

    // phase 4: seg_mean finalize + coord output
    rowscale<<<cdiv_i((long long)M * DIMF, 256), 256, 0, stream>>>(econv, cnt_ei, (long long)M * DIMF);
    rowscale<<<cdiv_i((long long)N * DIMF, 256), 256, 0, stream>>>(zconv, cnt_ni, (long long)N * DIMF);
    coord_finalize<<<cdiv_i((long long)N * 3, 256), 256, 0, stream>>>(coord, cacc, cnt_ni, out_c, N * 3);

    // phase 5: LN -> GELU -> residual -> z/e outputs + pooling accumulation
    ln_gelu_res_pool<<<N, DIMF, 0, stream>>>(zconv, x,  lng, lnb, nb, out_z, poolz, pcz, N);
    ln_gelu_res_pool<<<M, DIMF, 0, stream>>>(econv, he, lng, lnb, eb, out_e, poole, pce, M);

    // phase 6: GCN branches (reuse buf_xw / buf_ew)
    degree_accum<<<cdiv_i(EG, 256), 256, 0, stream>>>(XX + EG, deg_x, EG);
    degree_accum<<<cdiv_i(EE, 256), 256, 0, stream>>>(EEp + EE, deg_e, EE);
    gemm128_wmma<<<cdiv_i(N, 16), 256, 0, stream>>>(x,  Wx,  nullptr, buf_xw, N, DIMF);
    gemm128_wmma<<<cdiv_i(M, 16), 256, 0, stream>>>(he, Weg, nullptr, buf_ew, M, DIMF);
    gcn_scatter<<<EG, DIMF, 0, stream>>>(buf_xw, XX, XX + EG, deg_x, zimp, EG);
    gcn_scatter<<<EE, DIMF, 0, stream>>>(buf_ew, EEp, EEp + EE, deg_e, eimp, EE);
    gcn_finalize<<<cdiv_i((long long)N * DIMF, 256), 256, 0, stream>>>(zimp, buf_xw, deg_x, bx,  (long long)N * DIMF);
    gcn_finalize<<<cdiv_i((long long)M * DIMF, 256), 256, 0, stream>>>(eimp, buf_ew, deg_e, beg, (long long)M * DIMF);

    // phase 7: Barlow column stats
    colstats<<<cdiv_i(N, 256), DIMF, 0, stream>>>(zimp,  N, st + 0 * DIMF, st + 1 * DIMF);
    colstats<<<cdiv_i(N, 256), DIMF, 0, stream>>>(zconv, N, st + 2 * DIMF, st + 3 * DIMF);
    colstats<<<cdiv_i(M, 256), DIMF, 0, stream>>>(eimp,  M, st + 4 * DIMF, st + 5 * DIMF);
    colstats<<<cdiv_i(M, 256), DIMF, 0, stream>>>(econv, M, st + 6 * DIMF, st + 7 * DIMF);

    // phase 8: Gram matrices via split-K WMMA + atomics
    {
        dim3 g1(cdiv_i(N, 1024), 8), g2(cdiv_i(M, 1024), 8);
        corr_gram_wmma<<<g1, 256, 0, stream>>>(zimp, zconv, G1, N, 1024);
        corr_gram_wmma<<<g2, 256, 0, stream>>>(eimp, econv, G2, M, 1024);
    }

    // phase 9: Barlow losses -> scalar output
    barlow_loss<<<1, 256, 0, stream>>>(G1, st + 0 * DIMF, st + 1 * DIMF, st + 2 * DIMF, st + 3 * DIMF, N, out_l);
    barlow_loss<<<1, 256, 0, stream>>>(G2, st + 4 * DIMF, st + 5 * DIMF, st + 6 * DIMF, st + 7 * DIMF, M, out_l);

    // phase 10: pooled graph features + final projection [32,256]@[256,128]+bf
    pool_finalize<<<cdiv_i(NGRAPH * DIMF, 256), 256, 0, stream>>>(poolz, poole, pcz, pce, pooled);
    gemm128_wmma<<<cdiv_i(NGRAPH, 16), 256, 0, stream>>>(pooled, Wf, bf, out_g, NGRAPH, 2 * DIMF);
}